// Decoder_42417097016016
// MI455X (gfx1250) — compile-verified
//
#include <hip/hip_runtime.h>
#include <math.h>

// ---------------------------------------------------------------------------
// Shapes (fixed by the reference)
// ---------------------------------------------------------------------------
#define BATCH   16
#define T_DEC   512
#define T_ENC   256
#define DMODEL  256
#define F_DIM   400
#define ROWS    (BATCH * T_DEC)          // 8192
#define MEL_SZ  (ROWS * F_DIM)           // 3,276,800
#define ATTN_SZ (ROWS * T_ENC)           // 2,097,152

typedef __attribute__((ext_vector_type(2))) float v2f;
typedef __attribute__((ext_vector_type(8))) float v8f;

// D = A(16x4, f32) * B(4x16, f32) + C(16x16, f32)  -- full fp32 WMMA
__device__ __forceinline__ v8f wmma4(v2f a, v2f b, v8f c) {
    return __builtin_amdgcn_wmma_f32_16x16x4_f32(
        /*neg_a=*/false, a, /*neg_b=*/false, b,
        /*c_mod=*/(short)0, c, /*reuse_a=*/false, /*reuse_b=*/false);
}

// ---------------------------------------------------------------------------
// Generic WMMA GEMM:  Y[r][o] = act( (sum_c X[r][c]*W[o][c] + bias[o] + res)*ps )
//   W addressed with runtime strides (wso along o, wsc along c) and an
//   optional per-batch base offset (wb, batch = row/512) so the same kernel
//   does conv1x1, projections, q@K^T and attn@V.
//   One wave computes a 64(M) x 16(N) tile: 4 accumulators share one B
//   fragment per K-step -> ~6 VMEM per 4 WMMA.  blockDim.x = 128.
// act: 0 = none, 1 = relu, 2 = sigmoid
// ---------------------------------------------------------------------------
__global__ void gemm_nt_kernel(const float* __restrict__ X, int ldx, int C,
                               const float* __restrict__ W, long wso, long wsc, long wb,
                               const float* __restrict__ bias,
                               const float* __restrict__ res, int ldres,
                               float* __restrict__ Y, int ldy,
                               int ntiles, float postscale, int act)
{
    const int wave  = threadIdx.x >> 5;
    const int lane  = threadIdx.x & 31;
    const int ntile = blockIdx.x * (blockDim.x >> 5) + wave;
    if (ntile >= ntiles) return;                       // wave-uniform exit

    const int rowBase = blockIdx.y * 64;               // 64 rows per wave
    const int m    = lane & 15;                        // M row (A) / N col (B)
    const int klo  = (lane >> 4) * 2;                  // K pair held by lane
    const int nb   = ntile * 16;

    const float* __restrict__ xr0 = X + (long)(rowBase + m) * ldx + klo;
    const float* __restrict__ xr1 = xr0 + 16L * ldx;
    const float* __restrict__ xr2 = xr0 + 32L * ldx;
    const float* __restrict__ xr3 = xr0 + 48L * ldx;
    const float* __restrict__ wr  = W + (long)(rowBase >> 9) * wb
                                      + (long)(nb + m) * wso + (long)klo * wsc;

    v8f acc[4] = {{}, {}, {}, {}};
    if (wsc == 1) {                                    // contiguous B: b64 loads
        for (int c0 = 0; c0 < C; c0 += 4) {
            const v2f b  = *(const v2f*)(wr + c0);
            const v2f a0 = *(const v2f*)(xr0 + c0);
            const v2f a1 = *(const v2f*)(xr1 + c0);
            const v2f a2 = *(const v2f*)(xr2 + c0);
            const v2f a3 = *(const v2f*)(xr3 + c0);
            acc[0] = wmma4(a0, b, acc[0]);
            acc[1] = wmma4(a1, b, acc[1]);
            acc[2] = wmma4(a2, b, acc[2]);
            acc[3] = wmma4(a3, b, acc[3]);
        }
    } else {                                           // strided B (attn@V)
        for (int c0 = 0; c0 < C; c0 += 4) {
            v2f b; b.x = wr[(long)c0 * wsc]; b.y = wr[(long)(c0 + 1) * wsc];
            const v2f a0 = *(const v2f*)(xr0 + c0);
            const v2f a1 = *(const v2f*)(xr1 + c0);
            const v2f a2 = *(const v2f*)(xr2 + c0);
            const v2f a3 = *(const v2f*)(xr3 + c0);
            acc[0] = wmma4(a0, b, acc[0]);
            acc[1] = wmma4(a1, b, acc[1]);
            acc[2] = wmma4(a2, b, acc[2]);
            acc[3] = wmma4(a3, b, acc[3]);
        }
    }

    // C/D layout: VGPR i, lane l -> M = i + 8*(l>=16), N = l&15
    const int ncol = nb + (lane & 15);
    const float bv = bias ? bias[ncol] : 0.0f;
    const int mhi  = (lane >> 4) * 8;
#pragma unroll
    for (int j = 0; j < 4; ++j) {
#pragma unroll
        for (int i = 0; i < 8; ++i) {
            const int mr = rowBase + j * 16 + i + mhi;
            float v = acc[j][i] + bv;
            if (res) v += res[(long)mr * ldres + ncol];
            v *= postscale;
            if (act == 1)      v = fmaxf(v, 0.0f);
            else if (act == 2) v = 1.0f / (1.0f + expf(-v));
            Y[(long)mr * ldy + ncol] = v;
        }
    }
}

// ---------------------------------------------------------------------------
// Highway causal dilated conv (K=3), channels-last.
//   X,Y : (8192, 256)   W : (512, 256, 3)   bias : (512)
//   y[o] = sum_k sum_c X[b, t-(2-k)*dil, c] * W[o][c][k] + bias[o]
//   a = y[0:256], g = y[256:512];  out = sig(g)*a + (1-sig(g))*x
// One wave: 32 rows x (16 a-ch + 16 g-ch) = 4 accumulators; the two W
// fragments are reused across row subtiles, the two A fragments across a/g.
// Left padding handled by clamping row + multiplying A fragment by 0/1 mask
// (EXEC stays all-ones for WMMA).
// ---------------------------------------------------------------------------
__global__ void highway_kernel(const float* __restrict__ X,
                               const float* __restrict__ W,
                               const float* __restrict__ bias,
                               float* __restrict__ Y, int dil)
{
    const int wave  = threadIdx.x >> 5;
    const int lane  = threadIdx.x & 31;
    const int ntile = blockIdx.x * (blockDim.x >> 5) + wave;   // 16 tiles
    const int rowBase = blockIdx.y * 32;                       // 32 rows/wave
    const int bidx = rowBase >> 9;
    const int t0   = rowBase & 511;
    const int mi   = lane & 15;
    const int klo  = (lane >> 4) * 2;
    const int nb   = ntile * 16;
    const int n    = nb + mi;

    v8f aA0 = {}, aA1 = {}, aG0 = {}, aG1 = {};
#pragma unroll
    for (int k = 0; k < 3; ++k) {
        const int tm0 = t0 + mi - (2 - k) * dil;
        const int tm1 = tm0 + 16;
        const float msk0 = (tm0 >= 0) ? 1.0f : 0.0f;
        const float msk1 = (tm1 >= 0) ? 1.0f : 0.0f;
        const int tc0 = tm0 >= 0 ? tm0 : 0;
        const int tc1 = tm1 >= 0 ? tm1 : 0;
        const float* __restrict__ xr0 = X + ((long)(bidx * 512 + tc0)) * DMODEL + klo;
        const float* __restrict__ xr1 = X + ((long)(bidx * 512 + tc1)) * DMODEL + klo;
        const float* __restrict__ wa  = W + (long)n * 768 + (long)klo * 3 + k;
        const float* __restrict__ wg  = wa + 256L * 768;
        for (int c0 = 0; c0 < DMODEL; c0 += 4) {
            v2f a0 = *(const v2f*)(xr0 + c0); a0.x *= msk0; a0.y *= msk0;
            v2f a1 = *(const v2f*)(xr1 + c0); a1.x *= msk1; a1.y *= msk1;
            v2f ba; ba.x = wa[c0 * 3]; ba.y = wa[c0 * 3 + 3];
            v2f bg; bg.x = wg[c0 * 3]; bg.y = wg[c0 * 3 + 3];
            aA0 = wmma4(a0, ba, aA0);
            aA1 = wmma4(a1, ba, aA1);
            aG0 = wmma4(a0, bg, aG0);
            aG1 = wmma4(a1, bg, aG1);
        }
    }

    const int ncol = nb + (lane & 15);
    const float bA = bias[ncol];
    const float bG = bias[ncol + 256];
    const int mhi  = (lane >> 4) * 8;
#pragma unroll
    for (int i = 0; i < 8; ++i) {
        {
            const int mr = rowBase + i + mhi;
            const float av = aA0[i] + bA;
            const float gv = aG0[i] + bG;
            const float s  = 1.0f / (1.0f + expf(-gv));
            const float xv = X[(long)mr * DMODEL + ncol];
            Y[(long)mr * DMODEL + ncol] = s * av + (1.0f - s) * xv;
        }
        {
            const int mr = rowBase + 16 + i + mhi;
            const float av = aA1[i] + bA;
            const float gv = aG1[i] + bG;
            const float s  = 1.0f / (1.0f + expf(-gv));
            const float xv = X[(long)mr * DMODEL + ncol];
            Y[(long)mr * DMODEL + ncol] = s * av + (1.0f - s) * xv;
        }
    }
}

// ---------------------------------------------------------------------------
// Row softmax over 256 elements, in place.  One block of 256 threads per row.
// ---------------------------------------------------------------------------
__global__ void softmax_kernel(float* __restrict__ S)
{
    __shared__ float red[256];
    const long r  = blockIdx.x;
    const int tid = threadIdx.x;
    const float v = S[r * 256 + tid];
    red[tid] = v;
    __syncthreads();
    for (int s = 128; s > 0; s >>= 1) {
        if (tid < s) red[tid] = fmaxf(red[tid], red[tid + s]);
        __syncthreads();
    }
    const float mx = red[0];
    __syncthreads();
    const float e = expf(v - mx);
    red[tid] = e;
    __syncthreads();
    for (int s = 128; s > 0; s >>= 1) {
        if (tid < s) red[tid] += red[tid + s];
        __syncthreads();
    }
    S[r * 256 + tid] = e / red[0];
}

// ---------------------------------------------------------------------------
// cat[r][0:256] = Rq[r], cat[r][256:512] = Q[r]
// ---------------------------------------------------------------------------
__global__ void cat_kernel(const float* __restrict__ Rq,
                           const float* __restrict__ Q,
                           float* __restrict__ Out)
{
    const long i = (long)blockIdx.x * blockDim.x + threadIdx.x;   // 8192*256
    const long r = i >> 8;
    const int  c = (int)(i & 255);
    Out[r * 512 + c]       = Rq[i];
    Out[r * 512 + 256 + c] = Q[i];
}

// ---------------------------------------------------------------------------
// done[r] = sigmoid( Xs[r,:] . fcw + fcb )
// ---------------------------------------------------------------------------
__global__ void done_kernel(const float* __restrict__ Xs,
                            const float* __restrict__ fcw,
                            const float* __restrict__ fcb,
                            float* __restrict__ out)
{
    const int r = blockIdx.x * blockDim.x + threadIdx.x;
    if (r >= ROWS) return;
    float s = fcb[0];
    for (int c = 0; c < DMODEL; ++c) s += Xs[(long)r * DMODEL + c] * fcw[c];
    out[r] = 1.0f / (1.0f + expf(-s));
}

// ---------------------------------------------------------------------------
// Host-side orchestration
// ---------------------------------------------------------------------------
static void launch_gemm(const float* X, int ldx, int C,
                        const float* W, long wso, long wsc, long wb,
                        const float* bias, const float* res, int ldres,
                        float* Y, int ldy, int O, float ps, int act,
                        hipStream_t stream)
{
    const int ntiles = O / 16;
    dim3 blk(128);
    dim3 grd((ntiles + 3) / 4, ROWS / 64);
    gemm_nt_kernel<<<grd, blk, 0, stream>>>(X, ldx, C, W, wso, wsc, wb,
                                            bias, res, ldres, Y, ldy,
                                            ntiles, ps, act);
}

static void launch_highway(const float* X, const float* W, const float* b,
                           float* Y, int dil, hipStream_t stream)
{
    dim3 blk(128);
    dim3 grd(4, ROWS / 32);    // 16 n-tiles / 4 waves per block
    highway_kernel<<<grd, blk, 0, stream>>>(X, W, b, Y, dil);
}

extern "C" void kernel_launch(void* const* d_in, const int* in_sizes, int n_in,
                              void* d_out, int out_size, void* d_ws, size_t ws_size,
                              hipStream_t stream)
{
    const float* inputs  = (const float*)d_in[0];   // (16,512,400) channels-last
    const float* keys    = (const float*)d_in[1];   // (16,256,256)
    const float* values  = (const float*)d_in[2];   // (16,256,256)
    const float* enc_w0  = (const float*)d_in[3];
    const float* enc_b0  = (const float*)d_in[4];
    const float* enc_w1  = (const float*)d_in[5];
    const float* enc_b1  = (const float*)d_in[6];
    const float* enc_w2  = (const float*)d_in[7];
    const float* enc_b2  = (const float*)d_in[8];
    const float* enc_hw_w = (const float*)d_in[9];  // (10,512,256,3)
    const float* enc_hw_b = (const float*)d_in[10]; // (10,512)
    const float* attn_q_w = (const float*)d_in[11];
    const float* attn_q_b = (const float*)d_in[12];
    const float* attn_o_w = (const float*)d_in[13];
    const float* attn_o_b = (const float*)d_in[14];
    const float* dec_w0  = (const float*)d_in[15];  // (256,512,1)
    const float* dec_b0  = (const float*)d_in[16];
    const float* dec_hw_w = (const float*)d_in[17]; // (6,512,256,3)
    const float* dec_hw_b = (const float*)d_in[18]; // (6,512)
    const float* dec_w1  = (const float*)d_in[19];
    const float* dec_b1  = (const float*)d_in[20];
    const float* dec_w2  = (const float*)d_in[21];
    const float* dec_b2  = (const float*)d_in[22];
    const float* dec_w3  = (const float*)d_in[23];
    const float* dec_b3  = (const float*)d_in[24];
    const float* last_w  = (const float*)d_in[25];  // (400,256,1)
    const float* last_b  = (const float*)d_in[26];
    const float* fc_w    = (const float*)d_in[27];  // (1,256)
    const float* fc_b    = (const float*)d_in[28];

    float* mel  = (float*)d_out;                 // (8192,400)
    float* attn = mel + MEL_SZ;                  // (8192,256)
    float* done = attn + ATTN_SZ;                // (8192)

    float* bufA  = (float*)d_ws;                 // (8192,256)
    float* bufB  = bufA  + (long)ROWS * 256;
    float* bufC  = bufB  + (long)ROWS * 256;
    float* bufRq = bufC  + (long)ROWS * 256;
    float* cat   = bufRq + (long)ROWS * 256;     // (8192,512)

    static const int ENC_DIL[10] = {1, 3, 9, 27, 1, 3, 9, 27, 3, 3};
    static const int DEC_DIL[6]  = {1, 3, 9, 27, 1, 1};

    // ---- encoder prenet (conv1x1 stack) ----
    launch_gemm(inputs, F_DIM, F_DIM, enc_w0, F_DIM, 1, 0, enc_b0, nullptr, 0,
                bufA, DMODEL, DMODEL, 1.0f, /*relu*/1, stream);
    launch_gemm(bufA, DMODEL, DMODEL, enc_w1, DMODEL, 1, 0, enc_b1, nullptr, 0,
                bufB, DMODEL, DMODEL, 1.0f, 1, stream);
    launch_gemm(bufB, DMODEL, DMODEL, enc_w2, DMODEL, 1, 0, enc_b2, nullptr, 0,
                bufA, DMODEL, DMODEL, 1.0f, 0, stream);

    // ---- 10 encoder highway convs (ping-pong A<->B) ----
    float* src = bufA; float* dst = bufB;
    for (int i = 0; i < 10; ++i) {
        launch_highway(src, enc_hw_w + (long)i * 512 * 256 * 3,
                       enc_hw_b + (long)i * 512, dst, ENC_DIL[i], stream);
        float* t = src; src = dst; dst = t;
    }
    float* Q = src;                      // encoder output == query (8192,256)
    float* other = dst;                  // the free ping-pong buffer

    // ---- attention ----
    // q = query @ Wq^T + bq
    launch_gemm(Q, DMODEL, DMODEL, attn_q_w, DMODEL, 1, 0, attn_q_b, nullptr, 0,
                other, DMODEL, DMODEL, 1.0f, 0, stream);
    // scores = q @ keys^T  (per-batch W base), straight into d_out attn slot
    launch_gemm(other, DMODEL, DMODEL, keys, DMODEL, 1, (long)T_ENC * DMODEL,
                nullptr, nullptr, 0, attn, T_ENC, T_ENC, 1.0f, 0, stream);
    softmax_kernel<<<dim3(ROWS), dim3(256), 0, stream>>>(attn);
    // ctx = (attn @ values) * sqrt(T_enc)=16   (values: stride 1 along d, 256 along s)
    launch_gemm(attn, T_ENC, T_ENC, values, 1, DMODEL, (long)T_ENC * DMODEL,
                nullptr, nullptr, 0, bufC, DMODEL, DMODEL, 16.0f, 0, stream);
    // Rq = (ctx @ Wo^T + bo + query) * sqrt(0.5)
    launch_gemm(bufC, DMODEL, DMODEL, attn_o_w, DMODEL, 1, 0, attn_o_b,
                Q, DMODEL, bufRq, DMODEL, DMODEL, 0.70710678118654752f, 0, stream);

    // ---- decoder ----
    cat_kernel<<<dim3(ROWS), dim3(256), 0, stream>>>(bufRq, Q, cat);
    launch_gemm(cat, 2 * DMODEL, 2 * DMODEL, dec_w0, 2 * DMODEL, 1, 0, dec_b0,
                nullptr, 0, bufA, DMODEL, DMODEL, 1.0f, 0, stream);

    src = bufA; dst = bufB;
    for (int i = 0; i < 6; ++i) {
        launch_highway(src, dec_hw_w + (long)i * 512 * 256 * 3,
                       dec_hw_b + (long)i * 512, dst, DEC_DIL[i], stream);
        float* t = src; src = dst; dst = t;
    }
    // src now holds highway output
    launch_gemm(src, DMODEL, DMODEL, dec_w1, DMODEL, 1, 0, dec_b1, nullptr, 0,
                dst, DMODEL, DMODEL, 1.0f, 1, stream);
    launch_gemm(dst, DMODEL, DMODEL, dec_w2, DMODEL, 1, 0, dec_b2, nullptr, 0,
                src, DMODEL, DMODEL, 1.0f, 1, stream);
    launch_gemm(src, DMODEL, DMODEL, dec_w3, DMODEL, 1, 0, dec_b3, nullptr, 0,
                dst, DMODEL, DMODEL, 1.0f, 1, stream);
    float* dec_states = dst;             // (8192,256)

    // mel = sigmoid(dec_states @ last_w^T + last_b)  -> (8192,400) directly in d_out
    launch_gemm(dec_states, DMODEL, DMODEL, last_w, DMODEL, 1, 0, last_b,
                nullptr, 0, mel, F_DIM, F_DIM, 1.0f, /*sigmoid*/2, stream);

    // done = sigmoid(dec_states @ fc_w^T + fc_b)
    done_kernel<<<dim3(ROWS / 256), dim3(256), 0, stream>>>(dec_states, fc_w, fc_b, done);
}